// StandardGCN_58153857188499
// MI455X (gfx1250) — compile-verified
//
#include <hip/hip_runtime.h>
#include <hip/hip_bf16.h>

#define NN 200000
#define NE 3200000
#define NG 512
#define CH 16

typedef __attribute__((ext_vector_type(2))) float v2f;
typedef __attribute__((ext_vector_type(8))) float v8f;

__global__ void fill_zero_kernel(float* __restrict__ p, int n) {
  int i = blockIdx.x * blockDim.x + threadIdx.x;
  if (i < n) p[i] = 0.0f;
}

// Layer-1 aggregation: agg1[dst] += x[src]  (C_in = 1)
__global__ void edge_scatter1_kernel(const int* __restrict__ src, const int* __restrict__ dst,
                                     const float* __restrict__ x, float* __restrict__ agg1) {
  int e = blockIdx.x * blockDim.x + threadIdx.x;
  if (e >= NE) return;
  atomicAdd(&agg1[dst[e]], x[src[e]]);
}

// h1[i][c] = relu(agg1[i]*W_rel1[c] + x[i]*W_root1[c] + b_rel1[c])
__global__ void xform1_kernel(const float* __restrict__ x, const float* __restrict__ agg1,
                              const float* __restrict__ Wroot1, const float* __restrict__ Wrel1,
                              const float* __restrict__ brel1, float* __restrict__ h1) {
  int i = blockIdx.x * blockDim.x + threadIdx.x;
  if (i >= NN * CH) return;
  int node = i >> 4, c = i & 15;
  float v = agg1[node] * Wrel1[c] + x[node] * Wroot1[c] + brel1[c];
  h1[i] = v > 0.0f ? v : 0.0f;
}

// Layer-2 aggregation: one thread per (edge, channel) -> coalesced gather + coalesced atomics
__global__ void edge_scatter2_kernel(const int* __restrict__ src, const int* __restrict__ dst,
                                     const float* __restrict__ h1, float* __restrict__ agg2) {
  unsigned t = blockIdx.x * blockDim.x + threadIdx.x;
  if (t >= (unsigned)NE * CH) return;
  int e = (int)(t >> 4), c = (int)(t & 15);
  atomicAdd(&agg2[dst[e] * CH + c], h1[src[e] * CH + c]);
}

// h2 = relu(agg2 @ W_rel2 + h1 @ W_root2 + b_rel2), one wave per 16-node tile.
// Uses V_WMMA_F32_16X16X4_F32: A 16x4 f32 (2 VGPR/lane), B 4x16 f32, C/D 16x16 f32 (8 VGPR).
__global__ void xform2_wmma_kernel(const float* __restrict__ agg2, const float* __restrict__ h1,
                                   const float* __restrict__ Wrel2, const float* __restrict__ Wroot2,
                                   const float* __restrict__ brel2, float* __restrict__ h2) {
  const int wavesPerBlk = blockDim.x >> 5;
  int tile = blockIdx.x * wavesPerBlk + (threadIdx.x >> 5);
  if (tile >= NN / 16) return;           // uniform per wave -> EXEC all-ones inside
  int lane = threadIdx.x & 31;
  int row  = lane & 15;                  // A: M-row / B,C,D: N-column owned by this lane
  int hi   = lane >> 4;                  // lane half selects K pair (A/B) and M+8 (C/D)
  int node = tile * 16 + row;

#if defined(__gfx1250__) && __has_builtin(__builtin_amdgcn_wmma_f32_16x16x4_f32)
  v8f acc = {};
  const float* arow = agg2 + node * CH;  // A operand rows (M = node)
  const float* hrow = h1   + node * CH;
#pragma unroll
  for (int k = 0; k < 4; ++k) {          // K = 16 as 4 chunks of 4
    int kb = 4 * k + 2 * hi;             // ISA A layout: lanes>=16 hold K=+2,+3
    v2f a = *(const v2f*)(arow + kb);
    v2f b; b.x = Wrel2[(kb + 0) * CH + row]; b.y = Wrel2[(kb + 1) * CH + row];
    acc = __builtin_amdgcn_wmma_f32_16x16x4_f32(false, a, false, b, (short)0, acc, false, false);
  }
#pragma unroll
  for (int k = 0; k < 4; ++k) {
    int kb = 4 * k + 2 * hi;
    v2f a = *(const v2f*)(hrow + kb);
    v2f b; b.x = Wroot2[(kb + 0) * CH + row]; b.y = Wroot2[(kb + 1) * CH + row];
    acc = __builtin_amdgcn_wmma_f32_16x16x4_f32(false, a, false, b, (short)0, acc, false, false);
  }
  float bn = brel2[row];                 // C/D element N-index == row for every VGPR
#pragma unroll
  for (int r = 0; r < 8; ++r) {          // C/D VGPR r -> M = r (+8 for upper lanes)
    int m = r + 8 * hi;
    float v = acc[r] + bn;
    h2[(tile * 16 + m) * CH + row] = v > 0.0f ? v : 0.0f;
  }
#else
  // Portable fallback (also what the host pass parses): same output mapping, plain FMAs.
  for (int r = 0; r < 8; ++r) {
    int m = r + 8 * hi;
    int n = tile * 16 + m;
    float v = brel2[row];
    for (int k = 0; k < CH; ++k)
      v += agg2[n * CH + k] * Wrel2[k * CH + row] + h1[n * CH + k] * Wroot2[k * CH + row];
    h2[n * CH + row] = v > 0.0f ? v : 0.0f;
  }
#endif
}

// pooled[batch[i]][c] += h2[i][c]
__global__ void pool_kernel(const float* __restrict__ h2, const int* __restrict__ batch,
                            float* __restrict__ pooled) {
  int i = blockIdx.x * blockDim.x + threadIdx.x;
  if (i >= NN * CH) return;
  int node = i >> 4, c = i & 15;
  atomicAdd(&pooled[batch[node] * CH + c], h2[i]);
}

// out[g][k] = pooled[g] . W_cls[:,k] + b_cls[k]
__global__ void classify_kernel(const float* __restrict__ pooled, const float* __restrict__ Wcls,
                                const float* __restrict__ bcls, float* __restrict__ out) {
  int i = blockIdx.x * blockDim.x + threadIdx.x;
  if (i >= NG * 2) return;
  int g = i >> 1, k = i & 1;
  float v = bcls[k];
#pragma unroll
  for (int c = 0; c < CH; ++c) v += pooled[g * CH + c] * Wcls[c * 2 + k];
  out[i] = v;
}

extern "C" void kernel_launch(void* const* d_in, const int* in_sizes, int n_in,
                              void* d_out, int out_size, void* d_ws, size_t ws_size,
                              hipStream_t stream) {
  const float* x      = (const float*)d_in[0];
  const int*   ei     = (const int*)  d_in[1];   // [2, NE] flattened: row0 = src, row1 = dst
  const int*   batch  = (const int*)  d_in[2];
  const float* Wroot1 = (const float*)d_in[3];
  const float* Wrel1  = (const float*)d_in[4];
  const float* brel1  = (const float*)d_in[5];
  const float* Wroot2 = (const float*)d_in[6];
  const float* Wrel2  = (const float*)d_in[7];
  const float* brel2  = (const float*)d_in[8];
  const float* Wcls   = (const float*)d_in[9];
  const float* bcls   = (const float*)d_in[10];
  float* out = (float*)d_out;

  // Workspace layout (floats): agg1 | h1 | agg2 | h2 | pooled  (~39 MB, fits 192MB L2)
  float* ws     = (float*)d_ws;
  float* agg1   = ws;
  float* h1     = agg1 + NN;
  float* agg2   = h1   + (size_t)NN * CH;
  float* h2     = agg2 + (size_t)NN * CH;
  float* pooled = h2   + (size_t)NN * CH;

  const int* src = ei;
  const int* dst = ei + NE;

  fill_zero_kernel<<<(NN + 255) / 256, 256, 0, stream>>>(agg1, NN);
  fill_zero_kernel<<<(NN * CH + 255) / 256, 256, 0, stream>>>(agg2, NN * CH);
  fill_zero_kernel<<<(NG * CH + 255) / 256, 256, 0, stream>>>(pooled, NG * CH);

  edge_scatter1_kernel<<<(NE + 255) / 256, 256, 0, stream>>>(src, dst, x, agg1);
  xform1_kernel<<<(NN * CH + 255) / 256, 256, 0, stream>>>(x, agg1, Wroot1, Wrel1, brel1, h1);
  edge_scatter2_kernel<<<(NE * CH + 255) / 256, 256, 0, stream>>>(src, dst, h1, agg2);

  // 12500 tiles, 4 waves (128 threads) per block -> 3125 blocks exactly
  xform2_wmma_kernel<<<(NN / 16) / 4, 128, 0, stream>>>(agg2, h1, Wrel2, Wroot2, brel2, h2);

  pool_kernel<<<(NN * CH + 255) / 256, 256, 0, stream>>>(h2, batch, pooled);
  classify_kernel<<<(NG * 2 + 255) / 256, 256, 0, stream>>>(pooled, Wcls, bcls, out);
}